// PlaygroundModel_66365834658304
// MI455X (gfx1250) — compile-verified
//
#include <hip/hip_runtime.h>

// CDNA5 / gfx1250: wave32, packed-f32 VALU (VOP3P v_pk_*), trans co-execution.
// Bandwidth-bound: 12 B HBM traffic per row -> ~4.3 us floor at 23.3 TB/s.
// 16 rows/thread to amortize the one-time SGPR weight setup (~180 insts) to ~11/row.

typedef __attribute__((ext_vector_type(2))) float v2f;
typedef __attribute__((ext_vector_type(4))) float v4f;

static __device__ __forceinline__ v2f splat2(float s) { v2f r; r.x = s; r.y = s; return r; }

static __device__ __forceinline__ float fexp2(float v) {
#if defined(__HIP_DEVICE_COMPILE__) && __has_builtin(__builtin_amdgcn_exp2f)
    return __builtin_amdgcn_exp2f(v);      // v_exp_f32 (trans pipe)
#else
    return exp2f(v);
#endif
}

static __device__ __forceinline__ float frcp(float v) {
#if defined(__HIP_DEVICE_COMPILE__) && __has_builtin(__builtin_amdgcn_rcpf)
    return __builtin_amdgcn_rcpf(v);       // v_rcp_f32 (trans pipe)
#else
    return 1.0f / v;
#endif
}

// moth(h) = a*h*sigmoid(c*h); sigmoid(t) = 1/(1+exp(-t)); exp(-c*h) = exp2(nc*h), nc = -c*log2(e).
// Packed over a pair of rows: 4 v_pk_* ops + 2 exp + 2 rcp per 2 elements.
static __device__ __forceinline__ v2f moth2(v2f h, float a, float nc) {
    v2f t = h * nc;                        // v_pk_mul_f32
    v2f e; e.x = fexp2(t.x); e.y = fexp2(t.y);
    v2f den = e + 1.0f;                    // v_pk_add_f32
    v2f r; r.x = frcp(den.x); r.y = frcp(den.y);
    return (h * a) * r;                    // 2x v_pk_mul_f32
}

#define ITERS 4u   // 4 rows per iteration -> 16 rows per thread

__global__ __launch_bounds__(256) void moth_mlp(
    const float* __restrict__ x,
    const float* __restrict__ W1, const float* __restrict__ B1,
    const float* __restrict__ A1, const float* __restrict__ C1,
    const float* __restrict__ W2, const float* __restrict__ B2,
    const float* __restrict__ A2, const float* __restrict__ C2,
    const float* __restrict__ W3, const float* __restrict__ B3,
    float* __restrict__ out, unsigned nrows)
{
    // ---- wave-uniform parameter loads -> SGPRs (s_load), uniform fp folding ----
    const float NLOG2E = -1.4426950408889634f;
    float w10[8], w11[8], b1v[8], a1v[8], n1v[8];
    #pragma unroll
    for (int j = 0; j < 8; ++j) {
        w10[j] = W1[j];            // W1[0][j]
        w11[j] = W1[8 + j];        // W1[1][j]
        b1v[j] = B1[j];
        a1v[j] = A1[j];
        n1v[j] = C1[j] * NLOG2E;   // fold -log2(e) into c once
    }
    float w2v[8][4], b2v[4], a2v[4], n2v[4], w3v[4];
    #pragma unroll
    for (int k = 0; k < 4; ++k) {
        b2v[k] = B2[k];
        a2v[k] = A2[k];
        n2v[k] = C2[k] * NLOG2E;
        w3v[k] = W3[k];
        #pragma unroll
        for (int j = 0; j < 8; ++j) w2v[j][k] = W2[j * 4 + k];
    }
    const float b3v = B3[0];

    const v4f* __restrict__ xin = (const v4f*)x;     // one v4f = 2 rows of [x0, x1]
    v4f* __restrict__ out4 = (v4f*)out;

    // thread-iteration id: stride 256 between iterations => iteration address
    // deltas are compile-time constants (fold into global_load immediate offsets)
    const unsigned base = blockIdx.x * (256u * ITERS) + threadIdx.x;

    #pragma unroll
    for (unsigned i = 0; i < ITERS; ++i) {
        const unsigned gid = base + i * 256u;        // handles rows [4*gid, 4*gid+4)
        if (gid * 4u >= nrows) continue;

        // ---- 4 rows: 2x global_load_b128 in, 1x NT global_store_b128 out ----
        v4f q0 = xin[2u * gid];
        v4f q1 = xin[2u * gid + 1u];
        // gfx1250 global_prefetch_b8: this thread's data 2 iterations (16KB) ahead
        __builtin_prefetch(xin + 2u * (base + (i + 2u) * 256u), 0, 3);

        v4f res;
        #pragma unroll
        for (int p = 0; p < 2; ++p) {
            v4f q = p ? q1 : q0;
            v2f x0; x0.x = q.x; x0.y = q.z;          // x[:,0] for the row pair
            v2f x1; x1.x = q.y; x1.y = q.w;          // x[:,1]

            // layer 1: 2 -> 8   (16 pk-FMAs per row pair)
            v2f h[8];
            #pragma unroll
            for (int j = 0; j < 8; ++j) {
                v2f acc = splat2(b1v[j]);
                acc += x0 * w10[j];                  // v_pk_fma_f32
                acc += x1 * w11[j];
                h[j] = moth2(acc, a1v[j], n1v[j]);
            }

            // layer 2: 8 -> 4   (32 pk-FMAs per row pair)
            v2f g[4];
            #pragma unroll
            for (int k = 0; k < 4; ++k) {
                v2f acc = splat2(b2v[k]);
                #pragma unroll
                for (int j = 0; j < 8; ++j) acc += h[j] * w2v[j][k];
                g[k] = moth2(acc, a2v[k], n2v[k]);
            }

            // layer 3: 4 -> 1
            v2f o = splat2(b3v);
            #pragma unroll
            for (int k = 0; k < 4; ++k) o += g[k] * w3v[k];

            if (p == 0) { res.x = o.x; res.y = o.y; }
            else        { res.z = o.x; res.w = o.y; }
        }

        __builtin_nontemporal_store(res, out4 + gid);   // streamed once, never re-read
    }
}

extern "C" void kernel_launch(void* const* d_in, const int* in_sizes, int n_in,
                              void* d_out, int out_size, void* d_ws, size_t ws_size,
                              hipStream_t stream) {
    const float* x  = (const float*)d_in[0];
    const float* W1 = (const float*)d_in[1];
    const float* b1 = (const float*)d_in[2];
    const float* a1 = (const float*)d_in[3];
    const float* c1 = (const float*)d_in[4];
    const float* W2 = (const float*)d_in[5];
    const float* b2 = (const float*)d_in[6];
    const float* a2 = (const float*)d_in[7];
    const float* c2 = (const float*)d_in[8];
    const float* W3 = (const float*)d_in[9];
    const float* b3 = (const float*)d_in[10];
    float* out = (float*)d_out;

    const unsigned nrows   = (unsigned)out_size;                  // B = 8388608
    const unsigned threads = (nrows + 3u) / 4u;                   // 4 rows / thread-iter
    const unsigned rows_per_block = 256u * ITERS;                 // thread-iters per block
    const unsigned grid = (threads + rows_per_block - 1u) / rows_per_block;  // 2048 blocks

    moth_mlp<<<grid, 256, 0, stream>>>(x, W1, b1, a1, c1, W2, b2, a2, c2, W3, b3,
                                       out, nrows);
}